// VectorQuantizer_8916352107141
// MI455X (gfx1250) — compile-verified
//
#include <hip/hip_runtime.h>
#include <math.h>

typedef __attribute__((ext_vector_type(2))) float v2f;
typedef __attribute__((ext_vector_type(8))) float v8f;

#define K_CODES 512
#define D_DIM   30
#define D_PAD   32
#define NPAIRS  16                    // D_PAD / 2 interleaved k-pairs
#define TIME    1024
#define CCH     512
#define NROWS   (TIME * CCH)          // 524288 rows
#define EL2     528                   // pair-row stride: 528*2 % 64 banks == 32
#define OUT_Q_ELEMS (D_DIM * NROWS)   // 15728640

// Monotone float <-> uint32 order map (min over uint == min over float).
__device__ inline uint32_t mapf(float f) {
  uint32_t u = __float_as_uint(f);
  return (u & 0x80000000u) ? ~u : (u | 0x80000000u);
}
__device__ inline float unmapf(uint32_t u) {
  uint32_t v = (u & 0x80000000u) ? (u ^ 0x80000000u) : ~u;
  return __uint_as_float(v);
}

// ---------------------------------------------------------------------------
// Zero the workspace accumulators (counts histogram + loss accumulator).
// ---------------------------------------------------------------------------
__global__ void vq_zero(int* __restrict__ counts, float* __restrict__ loss) {
  int i = threadIdx.x;
  if (i < K_CODES) counts[i] = 0;
  if (i == 0) *loss = 0.0f;
}

// ---------------------------------------------------------------------------
// Main kernel: each wave owns 16 rows (16 consecutive t, fixed c) and scans
// all 512 codes in 16-wide tiles with V_WMMA_F32_16X16X4_F32.
// Accumulated score = ||e_n||^2 - 2 * x_m . e_n  (row norm added for loss).
// Argmin tracked as branchless u64 key = (orderedFloat(score)<<32) | n.
// ---------------------------------------------------------------------------
__global__ __launch_bounds__(256) void vq_main(
    const float* __restrict__ x,        // (30, 512, 1024)
    const float* __restrict__ emb,      // (512, 30)
    float* __restrict__ out_q,          // (30, 512, 1024)
    float* __restrict__ out_idx,        // (524288,) as float
    int* __restrict__ counts,
    float* __restrict__ loss_acc) {
  __shared__ float emb_lds[NPAIRS * EL2 * 2];  // [pair p][n][j] = emb[n][2p+j]
  __shared__ float cnorm[K_CODES];
  __shared__ int   widx[8][16];                // per-wave row -> code

  const int tid = threadIdx.x;

  // Stage codebook as interleaved k-pairs, zero-padded to D_PAD; code norms.
  for (int i = tid; i < NPAIRS * K_CODES; i += 256) {
    int p = i >> 9, n = i & 511;
    int d0 = 2 * p, d1 = d0 + 1;
    int base = (p * EL2 + n) * 2;
    emb_lds[base]     = (d0 < D_DIM) ? emb[n * D_DIM + d0] : 0.0f;
    emb_lds[base + 1] = (d1 < D_DIM) ? emb[n * D_DIM + d1] : 0.0f;
  }
  for (int n = tid; n < K_CODES; n += 256) {
    float s = 0.0f;
    for (int d = 0; d < D_DIM; ++d) { float e = emb[n * D_DIM + d]; s += e * e; }
    cnorm[n] = s;
  }
  __syncthreads();

  const int wave  = tid >> 5;
  const int lane  = tid & 31;
  const int lhalf = lane >> 4;   // 0 or 1
  const int lm    = lane & 15;

  const int rc = blockIdx.x * 8 + wave;  // row-chunk id, [0, 32768)
  const int c  = rc >> 6;                // channel
  const int t0 = (rc & 63) << 4;         // base time (16 consecutive t)

  // A tile: lane holds row m=lm; reg r of step s holds k = 4s + r + 2*lhalf.
  // Scale A by -2 so WMMA accumulates -2*x.e; keep x^2 for the row norm.
  v2f   a[8];
  float rn_part = 0.0f;
#pragma unroll
  for (int s = 0; s < 8; ++s) {
    int d0 = 4 * s + 2 * lhalf;
    int d1 = d0 + 1;
    float x0 = (d0 < D_DIM) ? x[(size_t)d0 * NROWS + (size_t)c * TIME + t0 + lm] : 0.0f;
    float x1 = (d1 < D_DIM) ? x[(size_t)d1 * NROWS + (size_t)c * TIME + t0 + lm] : 0.0f;
    rn_part += x0 * x0 + x1 * x1;
    a[s].x = -2.0f * x0;
    a[s].y = -2.0f * x1;
  }

  unsigned long long best[8];
#pragma unroll
  for (int r = 0; r < 8; ++r) best[r] = ~0ull;

  // Scan all 512 codes in 32 tiles of 16; unroll 2 -> two independent
  // 8-deep WMMA accumulation chains in flight.
#pragma unroll 2
  for (int ct = 0; ct < 32; ++ct) {
    int   n  = ct * 16 + lm;
    float cn = cnorm[n];
    v8f acc = {cn, cn, cn, cn, cn, cn, cn, cn};  // C init = ||e_n||^2
#pragma unroll
    for (int s = 0; s < 8; ++s) {
      // pair index p = 2s + lhalf holds k = 4s + 2*lhalf (+1): one b64 load.
      v2f b = *(const v2f*)&emb_lds[((2 * s + lhalf) * EL2 + n) * 2];
      acc = __builtin_amdgcn_wmma_f32_16x16x4_f32(
          false, a[s], false, b, (short)0, acc, false, false);
    }
#pragma unroll
    for (int r = 0; r < 8; ++r) {  // row m = r + 8*lhalf, col n
      unsigned long long key =
          ((unsigned long long)mapf(acc[r]) << 32) | (unsigned)n;
      best[r] = key < best[r] ? key : best[r];
    }
  }

  // Cross-lane argmin within each 16-lane half (lanes share M, differ in N).
#pragma unroll
  for (int r = 0; r < 8; ++r) {
    for (int off = 8; off >= 1; off >>= 1) {
      unsigned long long ob = __shfl_xor(best[r], off, 32);
      best[r] = ob < best[r] ? ob : best[r];
    }
  }

  // Loss: sum over 16 rows of (score + ||x||^2) = sum(score) + sum(rn).
  float s_best = 0.0f;
#pragma unroll
  for (int r = 0; r < 8; ++r) s_best += unmapf((uint32_t)(best[r] >> 32));
  s_best += __shfl_xor(s_best, 16, 32);            // half0 rows + half1 rows
  float s_rn = rn_part;
  for (int off = 16; off >= 1; off >>= 1) s_rn += __shfl_xor(s_rn, off, 32);
  if (lane == 0) atomicAdd(loss_acc, s_best + s_rn);

  // Lanes 0 and 16 publish 8 row results each: indices, counts, widx.
  if (lm == 0) {
#pragma unroll
    for (int r = 0; r < 8; ++r) {
      int m    = r + 8 * lhalf;
      int code = (int)(best[r] & 0xFFFFFFFFull);
      widx[wave][m] = code;
      atomicAdd(&counts[code], 1);
      out_idx[(size_t)(t0 + m) * CCH + c] = (float)code;
    }
  }
  __syncthreads();

  // Gather quantized rows from LDS codebook; coalesced 64B stores per half.
  int code_t = widx[wave][lm];  // code for t = t0 + lm
#pragma unroll
  for (int db = 0; db < 15; ++db) {
    int d = 2 * db + lhalf;     // covers d = 0..29
    out_q[(size_t)d * NROWS + (size_t)c * TIME + t0 + lm] =
        emb_lds[((d >> 1) * EL2 + code_t) * 2 + (d & 1)];
  }
}

// ---------------------------------------------------------------------------
// Finalize: vq_loss = sum(min dist)/(N*D); perplexity from code usage counts.
// ---------------------------------------------------------------------------
__global__ __launch_bounds__(512) void vq_final(
    const int* __restrict__ counts, const float* __restrict__ loss_acc,
    float* __restrict__ vq_loss_out, float* __restrict__ perp_out) {
  __shared__ float red[512];
  int i = threadIdx.x;
  float p = (float)counts[i] * (1.0f / (float)NROWS);
  red[i] = p * logf(p + 1e-10f);
  __syncthreads();
  for (int s = 256; s > 0; s >>= 1) {
    if (i < s) red[i] += red[i + s];
    __syncthreads();
  }
  if (i == 0) {
    *perp_out    = expf(-red[0]);
    *vq_loss_out = *loss_acc / (float)((size_t)NROWS * D_DIM);
  }
}

// ---------------------------------------------------------------------------
extern "C" void kernel_launch(void* const* d_in, const int* in_sizes, int n_in,
                              void* d_out, int out_size, void* d_ws, size_t ws_size,
                              hipStream_t stream) {
  const float* x   = (const float*)d_in[0];   // (30, 512, 1024) f32
  const float* emb = (const float*)d_in[1];   // (512, 30) f32

  float* f = (float*)d_out;
  // d_out layout (flat, return order): [vq_loss(1) | out(15728640) | perp(1) | idx(524288)]
  float* out_loss = f;
  float* out_q    = f + 1;
  float* out_perp = f + 1 + OUT_Q_ELEMS;
  float* out_idx  = f + 2 + OUT_Q_ELEMS;

  int*   counts = (int*)d_ws;
  float* loss   = (float*)((char*)d_ws + K_CODES * sizeof(int));

  vq_zero<<<1, 512, 0, stream>>>(counts, loss);
  vq_main<<<4096, 256, 0, stream>>>(x, emb, out_q, out_idx, counts, loss);
  vq_final<<<1, 512, 0, stream>>>(counts, loss, out_loss, out_perp);
}